// ChebNet_64991445123400
// MI455X (gfx1250) — compile-verified
//
#include <hip/hip_runtime.h>
#include <hip/hip_bf16.h>
#include <math.h>

// ---------------- problem constants ----------------
#define N_NODES  100000
#define N_EDGES  1600000
#define N_GRAPHS 100

typedef __attribute__((ext_vector_type(2))) float v2f;
typedef __attribute__((ext_vector_type(8))) float v8f;

// ---------------- small utility kernels ----------------
__global__ void zero_f32(float* __restrict__ p, int n) {
    int t = blockIdx.x * blockDim.x + threadIdx.x;
    if (t < n) p[t] = 0.0f;
}

__global__ void deg_count(const int* __restrict__ src, float* __restrict__ deg, int E) {
    int e = blockIdx.x * blockDim.x + threadIdx.x;
    if (e < E) atomicAdd(&deg[src[e]], 1.0f);
}

__global__ void node_prep(const float* __restrict__ deg,
                          const int* __restrict__ batch,
                          const float* __restrict__ lmax,
                          float* __restrict__ dinv,
                          float* __restrict__ sc,
                          float* __restrict__ diag,
                          int n) {
    int i = blockIdx.x * blockDim.x + threadIdx.x;
    if (i >= n) return;
    float d = deg[i];
    dinv[i] = (d > 0.0f) ? rsqrtf(fmaxf(d, 1e-12f)) : 0.0f;
    float s = 2.0f / lmax[batch[i]];
    sc[i]   = s;
    diag[i] = s - 1.0f;
}

__global__ void edge_weight(const int* __restrict__ src, const int* __restrict__ dst,
                            const float* __restrict__ sc, const float* __restrict__ dinv,
                            float* __restrict__ wedge, int E) {
    int e = blockIdx.x * blockDim.x + threadIdx.x;
    if (e >= E) return;
    int s = src[e], d = dst[e];
    wedge[e] = -sc[s] * dinv[s] * dinv[d];
}

// out[i,f] = coef * diag[i] * v[i,f] - (prev ? prev[i,f] : 0)
__global__ void lhat_init(const float* __restrict__ v,
                          const float* __restrict__ prev,   // may be null
                          const float* __restrict__ diag,
                          float* __restrict__ out,
                          int n, int lgF, float coef) {
    int t = blockIdx.x * blockDim.x + threadIdx.x;
    if (t >= (n << lgF)) return;
    int i = t >> lgF;
    float r = coef * diag[i] * v[t];
    if (prev) r -= prev[t];
    out[t] = r;
}

// out[dst,f] += coef * w_edge[e] * v[src,f]   (4 floats per thread)
__global__ void lhat_scatter(const float* __restrict__ v,
                             float* __restrict__ out,
                             const int* __restrict__ src, const int* __restrict__ dst,
                             const float* __restrict__ wedge,
                             int E, int lgF4, int F, float coef) {
    int t = blockIdx.x * blockDim.x + threadIdx.x;
    int F4 = 1 << lgF4;
    if (t >= E * F4) return;
    int e = t >> lgF4;
    int c = (t & (F4 - 1)) << 2;
    float w = coef * wedge[e];
    int s = src[e], d = dst[e];
    const float4 vv = *(const float4*)(v + (size_t)s * F + c);
    float* o = out + (size_t)d * F + c;
    atomicAdd(o + 0, w * vv.x);
    atomicAdd(o + 1, w * vv.y);
    atomicAdd(o + 2, w * vv.z);
    atomicAdd(o + 3, w * vv.w);
}

// ---------------- fused Chebyshev GEMM: out = tanh(sum_k Txk @ W[k] + b) ----------------
// One wave32 per 16x16 output tile.  A: 16x4 f32 (v2f), B: 4x16 f32 (v2f),
// C/D: 16x16 f32 (v8f) — V_WMMA_F32_16X16X4_F32.
// C/D layout: VGPR r -> row (tileM*16 + r + (lane>=16 ? 8:0)), col = tileM? no: col = lane&15.
__global__ __launch_bounds__(32) void cheb_gemm(
        const float* __restrict__ T0, const float* __restrict__ T1,
        const float* __restrict__ T2, const float* __restrict__ T3,
        const float* __restrict__ W,      // [4, F, Fo] row-major
        const float* __restrict__ bias,   // [Fo]
        float* __restrict__ out,          // [N_NODES, Fo]
        int nMtiles, int F, int Fo) {
    int tile  = blockIdx.x;
    int tileM = tile % nMtiles;
    int tileN = tile / nMtiles;
    int lane  = threadIdx.x;
    int half  = lane >> 4;        // 0 for lanes 0-15, 1 for lanes 16-31
    int lid   = lane & 15;
    int m     = tileM * 16 + lid; // A-fragment row for this lane
    int ncol  = tileN * 16 + lid; // B/C/D column for this lane

    v8f c = {};
    const float* Ts[4] = {T0, T1, T2, T3};
#pragma unroll
    for (int k = 0; k < 4; ++k) {
        const float* A = Ts[k] + (size_t)m * F;
        const float* B = W + (size_t)k * F * Fo + ncol;
        for (int kk = 0; kk < F; kk += 4) {
            int kh = kk + 2 * half;          // interleaved K halves across lane groups
            float2 av = *(const float2*)(A + kh);
            v2f a; a.x = av.x; a.y = av.y;
            v2f b; b.x = B[(size_t)kh * Fo]; b.y = B[(size_t)(kh + 1) * Fo];
            c = __builtin_amdgcn_wmma_f32_16x16x4_f32(
                    false, a, false, b, (short)0, c, false, false);
        }
    }

    float bv = bias[ncol];
    int mrow = tileM * 16 + (half ? 8 : 0);
#pragma unroll
    for (int r = 0; r < 8; ++r) {
        out[(size_t)(mrow + r) * Fo + ncol] = tanhf(c[r] + bv);
    }
}

// ---------------- pooling (batch is sorted -> chunked partial sums) ----------------
#define POOL_CHUNK 128
__global__ void pool_kernel(const float* __restrict__ h,
                            const int* __restrict__ batch,
                            float* __restrict__ pooled,   // [N_GRAPHS, 64]
                            int n) {
    int f = threadIdx.x;                 // 0..63
    int start = blockIdx.x * POOL_CHUNK;
    if (start >= n) return;
    int end = start + POOL_CHUNK; if (end > n) end = n;
    int curg = batch[start];
    float acc = 0.0f;
    for (int i = start; i < end; ++i) {
        int g = batch[i];
        if (g != curg) {
            atomicAdd(&pooled[(size_t)curg * 64 + f], acc);
            acc = 0.0f; curg = g;
        }
        acc += h[(size_t)i * 64 + f];
    }
    atomicAdd(&pooled[(size_t)curg * 64 + f], acc);
}

__global__ void fc_kernel(const float* __restrict__ pooled,
                          const float* __restrict__ Wfc,  // [64,10]
                          const float* __restrict__ bfc,  // [10]
                          float* __restrict__ out) {      // [N_GRAPHS,10]
    int t = blockIdx.x * blockDim.x + threadIdx.x;
    if (t >= N_GRAPHS * 10) return;
    int g = t / 10, o = t % 10;
    float acc = bfc[o];
#pragma unroll
    for (int f = 0; f < 64; ++f)
        acc += pooled[(size_t)g * 64 + f] * Wfc[(size_t)f * 10 + o];
    out[t] = tanhf(acc);
}

// ---------------- launch orchestration ----------------
static inline int cdiv(int a, int b) { return (a + b - 1) / b; }

static void run_lhat(const float* v, const float* prev, float* outb,
                     const float* diag, const int* src, const int* dst,
                     const float* wedge, int F, float coef, hipStream_t stream) {
    int lgF = (F == 64) ? 6 : 5;
    int lgF4 = lgF - 2;
    int nEl = N_NODES << lgF;
    lhat_init<<<cdiv(nEl, 256), 256, 0, stream>>>(v, prev, diag, outb, N_NODES, lgF, coef);
    int nT = N_EDGES << lgF4;
    lhat_scatter<<<cdiv(nT, 256), 256, 0, stream>>>(v, outb, src, dst, wedge,
                                                    N_EDGES, lgF4, F, coef);
}

static void run_layer(const float* x0, float* T1, float* T2, float* T3, float* H,
                      const float* diag, const int* src, const int* dst,
                      const float* wedge, const float* W, const float* b,
                      int F, int Fo, hipStream_t stream) {
    // Tx1 = lhat(Tx0)
    run_lhat(x0, nullptr, T1, diag, src, dst, wedge, F, 1.0f, stream);
    // Tx2 = 2*lhat(Tx1) - Tx0
    run_lhat(T1, x0, T2, diag, src, dst, wedge, F, 2.0f, stream);
    // Tx3 = 2*lhat(Tx2) - Tx1
    run_lhat(T2, T1, T3, diag, src, dst, wedge, F, 2.0f, stream);
    // H = tanh(sum_k Txk @ W[k] + b)
    int nMtiles = N_NODES / 16;                 // 6250 exactly
    int nBlocks = nMtiles * (Fo / 16);
    cheb_gemm<<<nBlocks, 32, 0, stream>>>(x0, T1, T2, T3, W, b, H, nMtiles, F, Fo);
}

extern "C" void kernel_launch(void* const* d_in, const int* in_sizes, int n_in,
                              void* d_out, int out_size, void* d_ws, size_t ws_size,
                              hipStream_t stream) {
    const float* x     = (const float*)d_in[0];
    const int*   eidx  = (const int*)d_in[1];
    const int*   batch = (const int*)d_in[2];
    const float* lmax  = (const float*)d_in[3];
    const float* W1    = (const float*)d_in[4];
    const float* b1    = (const float*)d_in[5];
    const float* W2    = (const float*)d_in[6];
    const float* b2    = (const float*)d_in[7];
    const float* W3    = (const float*)d_in[8];
    const float* b3    = (const float*)d_in[9];
    const float* Wfc   = (const float*)d_in[10];
    const float* bfc   = (const float*)d_in[11];
    float* out = (float*)d_out;

    const int* src = eidx;
    const int* dst = eidx + N_EDGES;

    // workspace layout (floats)
    float* ws     = (float*)d_ws;
    float* deg    = ws;                       // 100000
    float* dinv   = ws +   100000;            // 100000
    float* diag   = ws +   200000;            // 100000
    float* sc     = ws +   300000;            // 100000
    float* wedge  = ws +   400000;            // 1.6M
    float* T1     = ws +  2000000;            // 6.4M
    float* T2     = ws +  8400000;            // 6.4M
    float* T3     = ws + 14800000;            // 6.4M
    float* H1     = ws + 21200000;            // 6.4M
    float* H2     = ws + 27600000;            // 6.4M
    float* pooled = ws + 34000000;            // 6400

    // ---- graph preprocessing ----
    zero_f32<<<cdiv(N_NODES, 256), 256, 0, stream>>>(deg, N_NODES);
    deg_count<<<cdiv(N_EDGES, 256), 256, 0, stream>>>(src, deg, N_EDGES);
    node_prep<<<cdiv(N_NODES, 256), 256, 0, stream>>>(deg, batch, lmax, dinv, sc, diag, N_NODES);
    edge_weight<<<cdiv(N_EDGES, 256), 256, 0, stream>>>(src, dst, sc, dinv, wedge, N_EDGES);

    // ---- three Chebyshev conv layers ----
    run_layer(x,  T1, T2, T3, H1, diag, src, dst, wedge, W1, b1, 64, 32, stream);
    run_layer(H1, T1, T2, T3, H2, diag, src, dst, wedge, W2, b2, 32, 64, stream);
    run_layer(H2, T1, T2, T3, H1, diag, src, dst, wedge, W3, b3, 64, 64, stream);

    // ---- pool + FC ----
    zero_f32<<<cdiv(N_GRAPHS * 64, 256), 256, 0, stream>>>(pooled, N_GRAPHS * 64);
    pool_kernel<<<cdiv(N_NODES, POOL_CHUNK), 64, 0, stream>>>(H1, batch, pooled, N_NODES);
    fc_kernel<<<cdiv(N_GRAPHS * 10, 128), 128, 0, stream>>>(pooled, Wfc, bfc, out);
}